// GCPNet_75582834475160
// MI455X (gfx1250) — compile-verified
//
#include <hip/hip_runtime.h>
#include <hip/hip_bf16.h>
#include <math.h>

// Problem constants (match reference)
#define NN    2048
#define NCC   40
#define DD    64
#define GG    8
#define TMAXX 32

typedef float v2f __attribute__((ext_vector_type(2)));
typedef float v8f __attribute__((ext_vector_type(8)));

#if defined(__gfx1250__) && \
    __has_builtin(__builtin_amdgcn_global_load_async_to_lds_b32) && \
    __has_builtin(__builtin_amdgcn_s_wait_asynccnt)
#define USE_ASYNC_LDS 1
typedef __attribute__((address_space(1))) int* gptr_i32;
typedef __attribute__((address_space(3))) int* lptr_i32;
#else
#define USE_ASYNC_LDS 0
#endif

// ---------------------------------------------------------------------------
// Generic fp32 WMMA GEMM:  O = act( A(MxK) @ B(KxN) + bias )
//   - one wave = one 16x16 f32 tile via V_WMMA_F32_16X16X4_F32
//   - TRANSA is a template param -> no runtime branch in the K loop
//   - bounds handling by *index clamping* (no EXEC divergence, no masking:
//     an OOB A-row only affects an OOB output row, which is never stored;
//     an OOB B-col only affects its own never-stored column)
//   - K multiple of 4 required (40 / 64 / 2048 all qualify)
// ---------------------------------------------------------------------------
template <int TRANSA>
__global__ void wmma_gemm_f32(const float* __restrict__ A,
                              const float* __restrict__ B,
                              const float* __restrict__ bias,
                              float* __restrict__ O,
                              int M, int K, int Nd, int lda, int relu)
{
    const int wave = threadIdx.x >> 5;            // 8 waves / block
    const int lane = threadIdx.x & 31;
    const int ntn  = (Nd + 15) >> 4;
    const int ntm  = (M  + 15) >> 4;
    const int tile = blockIdx.x * 8 + wave;       // wave-uniform
    if (tile >= ntm * ntn) return;                // uniform exit: EXEC stays full

    const int mt = tile / ntn;
    const int nt = tile - mt * ntn;
    const int m0 = mt << 4;
    const int n0 = nt << 4;

    const int half = lane >> 4;                   // 0 -> K{0,1}, 1 -> K{2,3}
    const int l15  = lane & 15;
    const int am   = m0 + l15;                    // A row this lane supplies
    const int bn   = n0 + l15;                    // B col this lane supplies
    const int amc  = am < M  ? am : (M - 1);      // clamped (always-legal) indices
    const int bnc  = bn < Nd ? bn : (Nd - 1);

    v8f acc = {0.f, 0.f, 0.f, 0.f, 0.f, 0.f, 0.f, 0.f};

    const int kmain = K & ~15;                    // unroll-by-16 main body
    for (int k0 = 0; k0 < kmain; k0 += 16) {
        if (!TRANSA && (k0 + 80 < K))             // stream A ahead (global_prefetch_b8)
            __builtin_prefetch(&A[(size_t)amc * lda + k0 + 80], 0, 1);
#pragma unroll
        for (int u = 0; u < 4; ++u) {
            const int ka = k0 + u * 4 + half * 2;
            v2f a, b;
            if (TRANSA) {
                a.x = A[(size_t)ka       * lda + amc];
                a.y = A[(size_t)(ka + 1) * lda + amc];
            } else {
                a = *(const v2f*)&A[(size_t)amc * lda + ka];   // b64 load
            }
            b.x = B[(size_t)ka       * Nd + bnc];
            b.y = B[(size_t)(ka + 1) * Nd + bnc];
            acc = __builtin_amdgcn_wmma_f32_16x16x4_f32(
                      false, a, false, b, (short)0, acc, false, false);
        }
    }
    for (int k0 = kmain; k0 < K; k0 += 4) {       // K%16 tail (e.g. K=40)
        const int ka = k0 + half * 2;
        v2f a, b;
        if (TRANSA) {
            a.x = A[(size_t)ka       * lda + amc];
            a.y = A[(size_t)(ka + 1) * lda + amc];
        } else {
            a = *(const v2f*)&A[(size_t)amc * lda + ka];
        }
        b.x = B[(size_t)ka       * Nd + bnc];
        b.y = B[(size_t)(ka + 1) * Nd + bnc];
        acc = __builtin_amdgcn_wmma_f32_16x16x4_f32(
                  false, a, false, b, (short)0, acc, false, false);
    }

    // D layout: VGPR r, lanes 0-15 -> row m0+r, lanes 16-31 -> row m0+8+r
    if (bn < Nd) {
        const float bb = bias ? bias[bn] : 0.f;
#pragma unroll
        for (int r = 0; r < 8; ++r) {
            const int row = m0 + r + (half << 3);
            if (row < M) {
                float v = acc[r] + bb;
                if (relu) v = fmaxf(v, 0.f);
                O[(size_t)row * Nd + bn] = v;
            }
        }
    }
}

// ---------------------------------------------------------------------------
// Sequential LSTM recurrence. Gx = x@Wi + b precomputed ([T,256]).
// Single persistent workgroup; Wh (64KB) is staged into LDS (320KB/WGP) since
// it is re-read every one of the T sequential steps. h,c,g also in LDS.
// Dynamic shared memory: (64*256 + 64 + 64 + 256) floats = 67072 bytes.
// Preload uses GLOBAL_LOAD_ASYNC_TO_LDS_B32 + s_wait_asynccnt when available.
// ---------------------------------------------------------------------------
__global__ void lstm_seq(const float* __restrict__ Gx,
                         const float* __restrict__ Wh,   // [64, 256] row-major
                         float* __restrict__ Hout,
                         int T, int n_keep)
{
    extern __shared__ float smem[];
    float* Whs = smem;                       // 64*256
    float* h   = smem + DD * 4 * DD;         // 64
    float* c   = h + DD;                     // 64
    float* g   = c + DD;                     // 256
    const int tid = threadIdx.x;             // 256 threads

#if USE_ASYNC_LDS
    for (int i = tid; i < DD * 4 * DD; i += 256) {
        __builtin_amdgcn_global_load_async_to_lds_b32(
            (gptr_i32)(Wh + i), (lptr_i32)(Whs + i), 0, 0);
    }
    __builtin_amdgcn_s_wait_asynccnt(0);
#else
    for (int i = tid; i < DD * 4 * DD; i += 256) Whs[i] = Wh[i];
#endif
    if (tid < DD) { h[tid] = 0.f; c[tid] = 0.f; }
    __syncthreads();

    for (int t = 0; t < T; ++t) {
        float acc = Gx[(size_t)t * (4 * DD) + tid];
#pragma unroll
        for (int k = 0; k < DD; ++k)
            acc += h[k] * Whs[k * (4 * DD) + tid];
        g[tid] = acc;
        __syncthreads();
        if (tid < DD) {
            const float gi = g[tid];
            const float gf = g[DD + tid];
            const float gg = g[2 * DD + tid];
            const float go = g[3 * DD + tid];
            const float si = 1.f / (1.f + __expf(-gi));
            const float sf = 1.f / (1.f + __expf(-gf));
            const float so = 1.f / (1.f + __expf(-go));
            const float cn = sf * c[tid] + si * tanhf(gg);
            const float hn = so * tanhf(cn);
            c[tid] = cn;
            h[tid] = hn;
            if (t < n_keep) Hout[(size_t)t * DD + tid] = hn;
        }
        __syncthreads();
    }
}

// ---------------------------------------------------------------------------
// vote = H[2048,64] @ Wo[64,1] + bo
// ---------------------------------------------------------------------------
__global__ void vote_matvec(const float* __restrict__ H,
                            const float* __restrict__ Wo,
                            const float* __restrict__ bo,
                            float* __restrict__ vote, int M)
{
    const int m = blockIdx.x * blockDim.x + threadIdx.x;
    if (m < M) {
        float s = bo[0];
#pragma unroll 8
        for (int k = 0; k < DD; ++k) s += H[(size_t)m * DD + k] * Wo[k];
        vote[m] = s;
    }
}

// ---------------------------------------------------------------------------
// out[g] = sigmoid( sum(vote[seg g]) / count_g );  one block per group
// ---------------------------------------------------------------------------
__global__ void segsum_sigmoid(const float* __restrict__ vote,
                               const int* __restrict__ slice_idx,
                               float* __restrict__ out)
{
    __shared__ float sh[256];
    const int gidx = blockIdx.x;
    int start = 0;
    for (int i = 0; i < gidx; ++i) start += slice_idx[i];
    const int cnt = slice_idx[gidx];

    float s = 0.f;
    for (int i = threadIdx.x; i < cnt; i += blockDim.x) s += vote[start + i];
    sh[threadIdx.x] = s;
    __syncthreads();
    for (int off = 128; off > 0; off >>= 1) {
        if (threadIdx.x < off) sh[threadIdx.x] += sh[threadIdx.x + off];
        __syncthreads();
    }
    if (threadIdx.x == 0) {
        const float v = sh[0] / (float)cnt;
        out[gidx] = 1.f / (1.f + __expf(-v));
    }
}

// ---------------------------------------------------------------------------
extern "C" void kernel_launch(void* const* d_in, const int* in_sizes, int n_in,
                              void* d_out, int out_size, void* d_ws, size_t ws_size,
                              hipStream_t stream)
{
    (void)in_sizes; (void)n_in; (void)out_size; (void)ws_size;

    const float* M_vv     = (const float*)d_in[0];
    const float* M_vc     = (const float*)d_in[1];
    const float* V_in     = (const float*)d_in[2];
    const float* C_in     = (const float*)d_in[3];
    const int*   slice    = (const int*)  d_in[4];
    const float* mlpC_W   = (const float*)d_in[5];
    const float* mlpC_b   = (const float*)d_in[6];
    const float* mlpV_W   = (const float*)d_in[7];
    const float* mlpV_b   = (const float*)d_in[8];
    const float* vote_W   = (const float*)d_in[9];
    const float* vote_b   = (const float*)d_in[10];
    const float* vote_Wo  = (const float*)d_in[11];
    const float* vote_bo  = (const float*)d_in[12];
    const float* lstmv_Wi = (const float*)d_in[13];
    const float* lstmv_Wh = (const float*)d_in[14];
    const float* lstmv_b  = (const float*)d_in[15];
    const float* lstmc_Wi = (const float*)d_in[16];
    const float* lstmc_Wh = (const float*)d_in[17];
    const float* lstmc_b  = (const float*)d_in[18];

    // -------- workspace layout (floats) --------
    float* ws = (float*)d_ws;
    size_t off = 0;
    float* Va  = ws + off; off += (size_t)NN * DD;
    float* Vb  = ws + off; off += (size_t)NN * DD;
    float* Ca  = ws + off; off += (size_t)NCC * DD;
    float* Cb  = ws + off; off += (size_t)NCC * DD;
    float* Z1  = ws + off; off += (size_t)NCC * DD;
    float* Z2  = ws + off; off += (size_t)NCC * DD;
    float* Z3  = ws + off; off += (size_t)NCC * DD;
    float* Xv  = ws + off; off += (size_t)(2 * NN) * DD;
    float* Gx  = ws + off; off += (size_t)(2 * NN) * (4 * DD);
    float* Y1  = ws + off; off += (size_t)NN * DD;
    float* Y2  = ws + off; off += (size_t)NN * DD;
    float* Y3  = ws + off; off += (size_t)NN * DD;
    float* U   = ws + off; off += (size_t)NCC * DD;
    float* Gc  = ws + off; off += (size_t)NCC * (4 * DD);
    float* vt  = ws + off; off += (size_t)NN;

    auto gemm = [&](const float* A, const float* B, const float* bias, float* O,
                    int M, int K, int Nd, int lda, int transA, int relu) {
        const int tiles  = ((M + 15) / 16) * ((Nd + 15) / 16);
        const int blocks = (tiles + 7) / 8;
        if (transA)
            wmma_gemm_f32<1><<<blocks, 256, 0, stream>>>(A, B, bias, O, M, K, Nd,
                                                         lda, relu);
        else
            wmma_gemm_f32<0><<<blocks, 256, 0, stream>>>(A, B, bias, O, M, K, Nd,
                                                         lda, relu);
    };

    const size_t lstm_smem = (size_t)(DD * 4 * DD + DD + DD + 4 * DD) * sizeof(float);

    // initial state
    (void)hipMemcpyAsync(Va, V_in, (size_t)NN  * DD * sizeof(float),
                         hipMemcpyDeviceToDevice, stream);
    (void)hipMemcpyAsync(Ca, C_in, (size_t)NCC * DD * sizeof(float),
                         hipMemcpyDeviceToDevice, stream);

    float* Vcur = Va; float* Vnxt = Vb;
    float* Ccur = Ca; float* Cnxt = Cb;

    for (int t = 0; t < TMAXX; ++t) {
        // ---- mlpC(C) ----
        gemm(Ccur, mlpC_W,               mlpC_b,          Z1, NCC, DD, DD, DD, 0, 1);
        gemm(Z1,   mlpC_W + DD * DD,     mlpC_b + DD,     Z2, NCC, DD, DD, DD, 0, 1);
        gemm(Z2,   mlpC_W + 2 * DD * DD, mlpC_b + 2 * DD, Z3, NCC, DD, DD, DD, 0, 0);
        // ---- x_v = [ M_vv@V ; M_vc@mlpC(C) ] ----
        gemm(M_vv, Vcur, nullptr, Xv,                   NN, NN,  DD, NN,  0, 0);
        gemm(M_vc, Z3,   nullptr, Xv + (size_t)NN * DD, NN, NCC, DD, NCC, 0, 0);
        // ---- gate pre-projection + sequential LSTM over 2N nodes ----
        gemm(Xv, lstmv_Wi, lstmv_b, Gx, 2 * NN, DD, 4 * DD, DD, 0, 0);
        lstm_seq<<<1, 256, lstm_smem, stream>>>(Gx, lstmv_Wh, Vnxt, 2 * NN, NN);
        // ---- mlpV(old V) ----
        gemm(Vcur, mlpV_W,               mlpV_b,          Y1, NN, DD, DD, DD, 0, 1);
        gemm(Y1,   mlpV_W + DD * DD,     mlpV_b + DD,     Y2, NN, DD, DD, DD, 0, 1);
        gemm(Y2,   mlpV_W + 2 * DD * DD, mlpV_b + 2 * DD, Y3, NN, DD, DD, DD, 0, 0);
        // ---- U = M_vc^T @ mlpV(V) ----
        gemm(M_vc, Y3, nullptr, U, NCC, NN, DD, NCC, /*transA=*/1, 0);
        // ---- color LSTM ----
        gemm(U, lstmc_Wi, lstmc_b, Gc, NCC, DD, 4 * DD, DD, 0, 0);
        lstm_seq<<<1, 256, lstm_smem, stream>>>(Gc, lstmc_Wh, Cnxt, NCC, NCC);
        // ping-pong
        float* tmp;
        tmp = Vcur; Vcur = Vnxt; Vnxt = tmp;
        tmp = Ccur; Ccur = Cnxt; Cnxt = tmp;
    }

    // ---- vote head ----
    gemm(Vcur, vote_W,           vote_b,      Y1, NN, DD, DD, DD, 0, 1);
    gemm(Y1,   vote_W + DD * DD, vote_b + DD, Y2, NN, DD, DD, DD, 0, 1);
    vote_matvec<<<(NN + 255) / 256, 256, 0, stream>>>(Y2, vote_Wo, vote_bo, vt, NN);
    segsum_sigmoid<<<GG, 256, 0, stream>>>(vt, slice, (float*)d_out);
}